// SPDNetwork_1_52733608460727
// MI455X (gfx1250) — compile-verified
//
#include <hip/hip_runtime.h>
#include <math.h>

// ---------------------------------------------------------------------------
// SPDNet forward for MI455X (gfx1250, wave32).
// One wave per batch matrix; matrices live in LDS; stage 1-2 bilinear GEMMs use
// V_WMMA_F32_16X16X4_F32; eigensolves are wave-cooperative cyclic Jacobi.
// ---------------------------------------------------------------------------

typedef __attribute__((ext_vector_type(2))) float v2f;
typedef __attribute__((ext_vector_type(8))) float v8f;

#define LD        33      // padded LDS leading dim (odd => conflict-free cols)
#define WPB       4       // waves (matrices) per block
#define EPS_REC   1e-4f

// ---- WMMA f32 16x16x4 fragment helpers (layouts per CDNA5 ISA 7.12.2) ----
// A (16x4):  VGPR0: lanes0-15 K=0, lanes16-31 K=2 ; VGPR1: K=1 / K=3
// B (4x16):  VGPR0: lanes0-15 K=0 N=lane, lanes16-31 K=2 ; VGPR1: K=1 / K=3
// C (16x16): VGPR i: lanes0-15 M=i, lanes16-31 M=i+8 ; N=lane&15

template<int M, int N, int K, bool TA, bool TB>
__device__ __forceinline__ void mmul(const float* A, int lda,
                                     const float* Bm, int ldb,
                                     float* C, int ldc, int lane) {
  const int l15 = lane & 15;
  const int khalf = (lane >> 4) << 1;   // 0 or 2
#pragma unroll
  for (int mt = 0; mt < M; mt += 16) {
#pragma unroll
    for (int nt = 0; nt < N; nt += 16) {
      v8f acc = {};
#pragma unroll
      for (int kt = 0; kt < K; kt += 4) {
        const int ar = mt + l15;
        const int ak = kt + khalf;
        v2f a, b;
        if (TA) { a.x = A[ak * lda + ar];      a.y = A[(ak + 1) * lda + ar]; }
        else    { a.x = A[ar * lda + ak];      a.y = A[ar * lda + ak + 1];   }
        const int bc = nt + l15;
        const int bk = kt + khalf;
        if (TB) { b.x = Bm[bc * ldb + bk];     b.y = Bm[bc * ldb + bk + 1];  }
        else    { b.x = Bm[bk * ldb + bc];     b.y = Bm[(bk + 1) * ldb + bc];}
        acc = __builtin_amdgcn_wmma_f32_16x16x4_f32(
            false, a, false, b, (short)0, acc, false, false);
      }
      const int cc = nt + l15;
      const int cr = mt + ((lane >> 4) << 3);
#pragma unroll
      for (int i = 0; i < 8; ++i) C[(cr + i) * ldc + cc] = acc[i];
    }
  }
}

// ---- wave-cooperative cyclic Jacobi eigensolver on LDS (leading n x n) ----
__device__ __forceinline__ void jacobi(float* Mx, float* U, int n, int npad,
                                       int sweeps, int lane) {
  // init U = I over the padded region (pad must be zero for later WMMA)
  for (int idx = lane; idx < npad * npad; idx += 32) {
    int r = idx / npad, c = idx % npad;
    U[r * LD + c] = (r == c && r < n) ? 1.f : 0.f;
  }
#pragma unroll 1
  for (int sw = 0; sw < sweeps; ++sw) {
#pragma unroll 1
    for (int p = 0; p < n - 1; ++p) {
#pragma unroll 1
      for (int q = p + 1; q < n; ++q) {
        float app = Mx[p * LD + p];
        float aqq = Mx[q * LD + q];
        float apq = Mx[p * LD + q];
        float c, s;
        if (fabsf(apq) > 1e-30f) {
          float tau = (aqq - app) / (2.f * apq);
          float t = (tau >= 0.f ? 1.f : -1.f) /
                    (fabsf(tau) + sqrtf(1.f + tau * tau));
          c = 1.f / sqrtf(1.f + t * t);
          s = t * c;
        } else { c = 1.f; s = 0.f; }
        int j = lane;
        if (j < n) {   // rows p,q :  M <- J^T M
          float m1 = Mx[p * LD + j], m2 = Mx[q * LD + j];
          Mx[p * LD + j] = c * m1 - s * m2;
          Mx[q * LD + j] = s * m1 + c * m2;
        }
        if (j < n) {   // cols p,q :  M <- M J ;  U <- U J
          float m1 = Mx[j * LD + p], m2 = Mx[j * LD + q];
          Mx[j * LD + p] = c * m1 - s * m2;
          Mx[j * LD + q] = s * m1 + c * m2;
          float u1 = U[j * LD + p], u2 = U[j * LD + q];
          U[j * LD + p] = c * u1 - s * u2;
          U[j * LD + q] = s * u1 + c * u2;
        }
      }
    }
  }
}

// ---- T = U * diag(max(eig, EPS)); zero outside n x n so WMMA pads are clean
template<int NP, int N>
__device__ __forceinline__ void recbuild(const float* Mx, const float* U,
                                         float* T, int lane) {
#pragma unroll 1
  for (int idx = lane; idx < NP * NP; idx += 32) {
    int r = idx / NP, c = idx % NP;
    float dv = (c < N) ? fmaxf(Mx[c * LD + c], EPS_REC) : 0.f;
    T[r * LD + c] = (r < N) ? U[r * LD + c] * dv : 0.f;
  }
}

__global__ __launch_bounds__(WPB * 32)
void spdnet_fwd_kernel(const float* __restrict__ x,
                       const float* __restrict__ w1,
                       const float* __restrict__ w2,
                       const float* __restrict__ w3,
                       const float* __restrict__ fcw,
                       float* __restrict__ outLog,
                       float* __restrict__ outFeat, int Btot) {
  __shared__ float sW1[32 * 32];
  __shared__ float sW2[32 * 16];
  __shared__ float sW3[16 * 16];
  __shared__ float sBuf[WPB][3 * 32 * LD];

  const int tid = threadIdx.x;
  // block-cooperative zero-padded weight staging
  for (int idx = tid; idx < 32 * 32; idx += WPB * 32) {
    int r = idx >> 5, c = idx & 31;
    sW1[idx] = (r < 28 && c < 28) ? w1[r * 28 + c] : 0.f;
  }
  for (int idx = tid; idx < 32 * 16; idx += WPB * 32) {
    int r = idx >> 4, c = idx & 15;
    sW2[idx] = (r < 28 && c < 14) ? w2[r * 14 + c] : 0.f;
  }
  for (int idx = tid; idx < 16 * 16; idx += WPB * 32) {
    int r = idx >> 4, c = idx & 15;
    sW3[idx] = (r < 14 && c < 3) ? w3[r * 3 + c] : 0.f;
  }
  __syncthreads();

  const int wid = tid >> 5, lane = tid & 31;
  const int b = blockIdx.x * WPB + wid;
  if (b >= Btot) return;   // wave-uniform; no barriers after this point

  float* A0 = &sBuf[wid][0];
  float* A1 = A0 + 32 * LD;
  float* A2 = A1 + 32 * LD;

  // ---- load X (28x28) zero-padded to 32x32 ----
  const float* xb = x + (size_t)b * 784;
  for (int idx = lane; idx < 32 * 32; idx += 32) {
    int r = idx >> 5, c = idx & 31;
    A0[r * LD + c] = (r < 28 && c < 28) ? xb[r * 28 + c] : 0.f;
  }

  // ---- stage 1: M1 = w1^T X w1 ; ReEig ----
  mmul<32, 32, 32, false, false>(A0, LD, sW1, 32, A1, LD, lane); // T  = X  * w1
  mmul<32, 32, 32, true,  false>(sW1, 32, A1, LD, A0, LD, lane); // M1 = w1^T * T
  jacobi(A0, A2, 28, 32, 9, lane);
  recbuild<32, 28>(A0, A2, A1, lane);                            // A1 = U*diag(d')
  mmul<32, 32, 32, false, true >(A1, LD, A2, LD, A0, LD, lane);  // R1 = A1 * U^T

  // ---- stage 2: M2 = w2^T R1 w2 ; ReEig ----
  mmul<32, 16, 32, false, false>(A0, LD, sW2, 16, A1, LD, lane); // T2 = R1 * w2
  mmul<16, 16, 32, true,  false>(sW2, 16, A1, LD, A0, LD, lane); // M2 = w2^T * T2
  jacobi(A0, A2, 14, 16, 8, lane);
  recbuild<16, 14>(A0, A2, A1, lane);
  mmul<16, 16, 16, false, true >(A1, LD, A2, LD, A0, LD, lane);  // R2 = A1 * U^T

  // ---- stage 3: M3 = w3^T R2 w3 (3x3, scalar per-lane) ; LogEig ----
  for (int idx = lane; idx < 42; idx += 32) {     // T3 = R2 * w3  (14x3)
    int j = idx / 3, i = idx % 3;
    float s = 0.f;
#pragma unroll 1
    for (int k = 0; k < 14; ++k) s += A0[j * LD + k] * sW3[k * 16 + i];
    A1[j * LD + i] = s;
  }
  if (lane < 9) {                                 // M3 = w3^T * T3  (3x3)
    int i = lane / 3, l = lane % 3;
    float s = 0.f;
#pragma unroll 1
    for (int j = 0; j < 14; ++j) s += sW3[j * 16 + i] * A1[j * LD + l];
    A0[i * LD + l] = s;
  }
  jacobi(A0, A2, 3, 3, 6, lane);

  // X3 = U diag(log d) U^T -> feat[9]
  if (lane < 9) {
    int i = lane / 3, j = lane % 3;
    float s = 0.f;
#pragma unroll
    for (int k = 0; k < 3; ++k)
      s += A2[i * LD + k] * logf(fmaxf(A0[k * LD + k], 1e-37f)) * A2[j * LD + k];
    A1[lane] = s;
    outFeat[(size_t)b * 9 + lane] = s;
  }
  // logits = feat @ fc_w  (9x3), then log_softmax
  if (lane < 3) {
    float l = 0.f;
#pragma unroll
    for (int f = 0; f < 9; ++f) l += A1[f] * fcw[f * 3 + lane];
    A1[16 + lane] = l;
  }
  if (lane < 3) {
    float l0 = A1[16], l1 = A1[17], l2 = A1[18];
    float mx = fmaxf(l0, fmaxf(l1, l2));
    float lse = logf(expf(l0 - mx) + expf(l1 - mx) + expf(l2 - mx));
    outLog[(size_t)b * 3 + lane] = A1[16 + lane] - mx - lse;
  }
}

extern "C" void kernel_launch(void* const* d_in, const int* in_sizes, int n_in,
                              void* d_out, int out_size, void* d_ws, size_t ws_size,
                              hipStream_t stream) {
  const float* x   = (const float*)d_in[0];
  const float* w1  = (const float*)d_in[1];
  const float* w2  = (const float*)d_in[2];
  const float* w3  = (const float*)d_in[3];
  const float* fcw = (const float*)d_in[4];
  const int B = in_sizes[0] / (28 * 28);

  float* outLog  = (float*)d_out;                 // [B,3] first (return order)
  float* outFeat = outLog + (size_t)B * 3;        // [B,9] second

  const int blocks = (B + WPB - 1) / WPB;
  spdnet_fwd_kernel<<<blocks, WPB * 32, 0, stream>>>(x, w1, w2, w3, fcw,
                                                     outLog, outFeat, B);
}